// PromptEmbedding_26328149524507
// MI455X (gfx1250) — compile-verified
//
#include <hip/hip_runtime.h>

typedef float v2f __attribute__((ext_vector_type(2)));
typedef float v4f __attribute__((ext_vector_type(4)));
typedef float v8f __attribute__((ext_vector_type(8)));

#define D_    256
#define H_    8
#define DH_   32
#define PLEN_ 10
#define TOK_  16          // tokens per workgroup tile
#define QS_LD 260         // LDS row stride (floats): bank-stride 4, 16B-aligned rows
#define HB_LD 132

// Dynamic LDS layout (floats)
#define OFF_QS   0                         // [16][260] queries tile -> later y/LN buffer
#define OFF_QP   (OFF_QS + TOK_*QS_LD)     // [16][260] qproj tile -> later ctx tile
#define OFF_SK   (OFF_QP + TOK_*QS_LD)     // [3*10*256] blended-K basis
#define OFF_SV   (OFF_SK + 3*PLEN_*D_)     // [3*10*256] blended-V basis
#define OFF_HB   (OFF_SV + 3*PLEN_*D_)     // [16][132] relu(h) tile
#define OFF_PW   (OFF_HB + TOK_*HB_LD)     // [16][3] prompt weights
#define OFF_LG   (OFF_PW + TOK_*3)         // [16][3] logits
#define OFF_RS   (OFF_LG + TOK_*3)         // [16][16] LN sum partials
#define OFF_RQ   (OFF_RS + TOK_*16)        // [16][16] LN sumsq partials
#define OFF_ST   (OFF_RQ + TOK_*16)        // [16][2]  LN mu/rstd
#define SMEM_FLOATS (OFF_ST + TOK_*2)      // total = 26432 floats = 105728 B

// ---------------------------------------------------------------------------
// K0: precompute Kt[j][p][e], Vt[j][p][e] = tables[j][p]·W{k,v}ᵀ + b{k,v}
// (valid because softmax weights sum to 1, so the bias folds in)
// grid: 30 blocks (j*10+p), 256 threads (e)
// ---------------------------------------------------------------------------
__global__ __launch_bounds__(256) void k_prep(
    const float* __restrict__ cp, const float* __restrict__ wp,
    const float* __restrict__ lp, const float* __restrict__ ipw,
    const float* __restrict__ ipb, float* __restrict__ Kt, float* __restrict__ Vt)
{
    __shared__ float Ts[D_];
    const int jp = blockIdx.x;
    const int j  = jp / PLEN_;
    const int p  = jp % PLEN_;
    const float* T = (j == 0 ? cp : (j == 1 ? wp : lp)) + (size_t)p * D_;
    Ts[threadIdx.x] = T[threadIdx.x];
    __syncthreads();
    const int e = threadIdx.x;
    const float* wk = ipw + (size_t)D_ * D_     + (size_t)e * D_;
    const float* wv = ipw + (size_t)2 * D_ * D_ + (size_t)e * D_;
    float ak = ipb[D_ + e];
    float av = ipb[2 * D_ + e];
    #pragma unroll 4
    for (int d = 0; d < D_; ++d) { float tv = Ts[d]; ak = fmaf(tv, wk[d], ak); av = fmaf(tv, wv[d], av); }
    Kt[(size_t)jp * D_ + e] = ak;
    Vt[(size_t)jp * D_ + e] = av;
}

// ---------------------------------------------------------------------------
// Fused per-16-token kernel: gating MLP -> softmax(3) -> Q proj -> attention
// over 10 prompt slots -> out proj -> residual -> LayerNorm.
// 256 threads = 8 waves. All GEMMs via V_WMMA_F32_16X16X4_F32, A from LDS.
// ---------------------------------------------------------------------------
__global__ __launch_bounds__(256) void k_fused(
    const float* __restrict__ q,   const float* __restrict__ w1,
    const float* __restrict__ b1,  const float* __restrict__ w2,
    const float* __restrict__ b2,  const float* __restrict__ wq,
    const float* __restrict__ bq,  const float* __restrict__ wo,
    const float* __restrict__ bo,  const float* __restrict__ g,
    const float* __restrict__ bb,  const float* __restrict__ Kt,
    const float* __restrict__ Vt,  float* __restrict__ out,
    float* __restrict__ pw_out)
{
    extern __shared__ float smem[];
    float* Qs  = smem + OFF_QS;   // queries tile (later: y buffer)
    float* qpS = smem + OFF_QP;   // qproj tile (later: ctx tile)
    float* sK  = smem + OFF_SK;
    float* sV  = smem + OFF_SV;
    float* hb  = smem + OFF_HB;
    float* pwS = smem + OFF_PW;
    float* lgS = smem + OFF_LG;
    float* rS  = smem + OFF_RS;
    float* rQ  = smem + OFF_RQ;
    float* st  = smem + OFF_ST;

    const int tid  = threadIdx.x;
    const int t0   = blockIdx.x * TOK_;
    const int wave = tid >> 5;
    const int lane = tid & 31;
    const int m    = lane & 15;
    const int koff = (lane >> 4) << 1;   // f32 WMMA: VGPR0 holds K = koff, VGPR1 K = koff+1
    const int hi8  = (lane >> 4) << 3;   // D rows: r (lanes 0-15) / r+8 (lanes 16-31)

    // ---- Phase A: stage queries tile + Kt/Vt into LDS (float4 loads) ----
    for (int i = tid; i < TOK_ * (D_ / 4); i += 256) {         // 1024 float4s
        int r = i >> 6, c4 = (i & 63) << 2;
        *(v4f*)&Qs[r * QS_LD + c4] = *(const v4f*)&q[(size_t)(t0 + r) * D_ + c4];
    }
    for (int i = tid; i < (3 * PLEN_ * D_) / 4; i += 256) {    // 1920 float4s each
        *(v4f*)&sK[i << 2] = *(const v4f*)&Kt[(size_t)i << 2];
        *(v4f*)&sV[i << 2] = *(const v4f*)&Vt[(size_t)i << 2];
    }
    __syncthreads();

    const float* arow = Qs + m * QS_LD + koff;   // WMMA A operand rows (LDS)

    // ---- Phase B: h = relu(q·W1ᵀ + b1), N=128: 8 waves x 16 cols ----
    {
        const int n0 = wave * 16;
        v8f c = {0.f,0.f,0.f,0.f,0.f,0.f,0.f,0.f};
        const float* brow = w1 + (size_t)(n0 + m) * D_ + koff;
        #pragma unroll 8
        for (int k0 = 0; k0 < D_; k0 += 4) {
            v2f a = *(const v2f*)(arow + k0);
            v2f b = *(const v2f*)(brow + k0);
            c = __builtin_amdgcn_wmma_f32_16x16x4_f32(false, a, false, b, (short)0, c, false, false);
        }
        const int col = n0 + m;
        const float bias = b1[col];
        #pragma unroll
        for (int r = 0; r < 8; ++r) {
            float v = c[r] + bias;
            hb[(r + hi8) * HB_LD + col] = v > 0.f ? v : 0.f;
        }
    }

    // ---- Phase C: qp = q·Wqᵀ + bq, N=256: 8 waves x 2 tiles ----
    {
        const int n0 = wave * 32, n1 = n0 + 16;
        v8f c0 = {0.f,0.f,0.f,0.f,0.f,0.f,0.f,0.f};
        v8f c1 = c0;
        const float* brow0 = wq + (size_t)(n0 + m) * D_ + koff;
        const float* brow1 = wq + (size_t)(n1 + m) * D_ + koff;
        #pragma unroll 8
        for (int k0 = 0; k0 < D_; k0 += 4) {
            v2f a  = *(const v2f*)(arow  + k0);
            v2f b0 = *(const v2f*)(brow0 + k0);
            v2f b1v = *(const v2f*)(brow1 + k0);
            c0 = __builtin_amdgcn_wmma_f32_16x16x4_f32(false, a, false, b0,  (short)0, c0, false, false);
            c1 = __builtin_amdgcn_wmma_f32_16x16x4_f32(false, a, false, b1v, (short)0, c1, false, false);
        }
        const int col0 = n0 + m, col1 = n1 + m;
        const float g0 = bq[col0], g1 = bq[col1];
        #pragma unroll
        for (int r = 0; r < 8; ++r) {
            int row = r + hi8;
            qpS[row * QS_LD + col0] = c0[r] + g0;
            qpS[row * QS_LD + col1] = c1[r] + g1;
        }
    }
    __syncthreads();

    // ---- Phase D: gating logits + softmax(3) -> prompt weights ----
    if (tid < TOK_ * 3) {
        int tt = tid / 3, j = tid % 3;
        float acc = b2[j];
        const float* w2r = w2 + j * 128;
        const float* hr  = hb + tt * HB_LD;
        #pragma unroll 4
        for (int o = 0; o < 128; ++o) acc = fmaf(hr[o], w2r[o], acc);
        lgS[tt * 3 + j] = acc;
    }
    __syncthreads();
    if (tid < TOK_) {
        int tt = tid;
        float l0 = lgS[tt*3+0], l1 = lgS[tt*3+1], l2 = lgS[tt*3+2];
        float mx = fmaxf(l0, fmaxf(l1, l2));
        float e0 = __expf(l0 - mx), e1 = __expf(l1 - mx), e2 = __expf(l2 - mx);
        float inv = 1.f / (e0 + e1 + e2);
        float p0 = e0 * inv, p1 = e1 * inv, p2 = e2 * inv;
        pwS[tt*3+0] = p0; pwS[tt*3+1] = p1; pwS[tt*3+2] = p2;
        size_t base = (size_t)(t0 + tt) * 3;
        pw_out[base+0] = p0; pw_out[base+1] = p1; pw_out[base+2] = p2;
    }
    __syncthreads();

    // ---- Phase E: attention (8 waves = 8 heads, lane = channel) ----
    {
        const int e = wave * DH_ + lane;
        const float inv_sqrt_dh = 0.17677669529663687f; // 1/sqrt(32)
        for (int tt = 0; tt < TOK_; ++tt) {
            const float p0 = pwS[tt*3+0], p1 = pwS[tt*3+1], p2 = pwS[tt*3+2];
            const float qv = qpS[tt * QS_LD + e];
            float sc[PLEN_], vv[PLEN_];
            #pragma unroll
            for (int p = 0; p < PLEN_; ++p) {
                const int idx = p * D_ + e;
                float kk = p0 * sK[idx] + p1 * sK[2560 + idx] + p2 * sK[5120 + idx];
                vv[p]    = p0 * sV[idx] + p1 * sV[2560 + idx] + p2 * sV[5120 + idx];
                float s = qv * kk;
                s += __shfl_xor(s, 16);
                s += __shfl_xor(s, 8);
                s += __shfl_xor(s, 4);
                s += __shfl_xor(s, 2);
                s += __shfl_xor(s, 1);
                sc[p] = s * inv_sqrt_dh;
            }
            float mx = sc[0];
            #pragma unroll
            for (int p = 1; p < PLEN_; ++p) mx = fmaxf(mx, sc[p]);
            float sum = 0.f, ctxv = 0.f;
            #pragma unroll
            for (int p = 0; p < PLEN_; ++p) {
                float ex = __expf(sc[p] - mx);
                sum  += ex;
                ctxv += ex * vv[p];
            }
            qpS[tt * QS_LD + e] = ctxv / sum;   // ctx overwrites qp slot (same owner thread)
        }
    }
    __syncthreads();

    // ---- Phase F: attn_out = ctx·Woᵀ + bo; y = queries + attn_out (into Qs) ----
    {
        const int n0 = wave * 32, n1 = n0 + 16;
        v8f c0 = {0.f,0.f,0.f,0.f,0.f,0.f,0.f,0.f};
        v8f c1 = c0;
        const float* crow  = qpS + m * QS_LD + koff;  // ctx is the A operand now
        const float* brow0 = wo + (size_t)(n0 + m) * D_ + koff;
        const float* brow1 = wo + (size_t)(n1 + m) * D_ + koff;
        #pragma unroll 8
        for (int k0 = 0; k0 < D_; k0 += 4) {
            v2f a  = *(const v2f*)(crow  + k0);
            v2f b0 = *(const v2f*)(brow0 + k0);
            v2f b1v = *(const v2f*)(brow1 + k0);
            c0 = __builtin_amdgcn_wmma_f32_16x16x4_f32(false, a, false, b0,  (short)0, c0, false, false);
            c1 = __builtin_amdgcn_wmma_f32_16x16x4_f32(false, a, false, b1v, (short)0, c1, false, false);
        }
        const int col0 = n0 + m, col1 = n1 + m;
        const float o0 = bo[col0], o1 = bo[col1];
        #pragma unroll
        for (int r = 0; r < 8; ++r) {
            int row = r + hi8;
            float* y0 = &Qs[row * QS_LD + col0];
            float* y1 = &Qs[row * QS_LD + col1];
            *y0 = c0[r] + o0 + *y0;            // residual: y = queries + attn_out, in place
            *y1 = c1[r] + o1 + *y1;
        }
    }
    __syncthreads();

    // ---- Phase G: LayerNorm over Qs rows, write final output ----
    {
        const int tt  = tid >> 4;
        const int seg = tid & 15;
        float s = 0.f, s2 = 0.f;
        #pragma unroll
        for (int c = 0; c < 16; ++c) {
            float y = Qs[tt * QS_LD + seg * 16 + c];
            s += y; s2 = fmaf(y, y, s2);
        }
        rS[tt * 16 + seg] = s;
        rQ[tt * 16 + seg] = s2;
        __syncthreads();
        if (tid < TOK_) {
            float a = 0.f, b = 0.f;
            #pragma unroll
            for (int i = 0; i < 16; ++i) { a += rS[tid * 16 + i]; b += rQ[tid * 16 + i]; }
            float mu  = a * (1.f / D_);
            float var = b * (1.f / D_) - mu * mu;
            st[tid * 2 + 0] = mu;
            st[tid * 2 + 1] = rsqrtf(var + 1e-5f);
        }
        __syncthreads();
        const float mu = st[tt * 2 + 0], rstd = st[tt * 2 + 1];
        size_t obase = (size_t)(t0 + tt) * D_;
        #pragma unroll
        for (int c = 0; c < 16; ++c) {
            int e = seg * 16 + c;
            out[obase + e] = (Qs[tt * QS_LD + e] - mu) * rstd * g[e] + bb[e];
        }
    }
}

// ---------------------------------------------------------------------------
extern "C" void kernel_launch(void* const* d_in, const int* in_sizes, int n_in,
                              void* d_out, int out_size, void* d_ws, size_t ws_size,
                              hipStream_t stream) {
    const float* q   = (const float*)d_in[0];
    const float* cp  = (const float*)d_in[1];
    const float* wp  = (const float*)d_in[2];
    const float* lp  = (const float*)d_in[3];
    const float* w1  = (const float*)d_in[4];
    const float* b1  = (const float*)d_in[5];
    const float* w2  = (const float*)d_in[6];
    const float* b2  = (const float*)d_in[7];
    const float* ipw = (const float*)d_in[8];
    const float* ipb = (const float*)d_in[9];
    const float* ow  = (const float*)d_in[10];
    const float* ob  = (const float*)d_in[11];
    const float* lng = (const float*)d_in[12];
    const float* lnb = (const float*)d_in[13];

    const int nTok = in_sizes[0] / D_;            // 32768
    const int nBlk = nTok / TOK_;                 // 2048

    float* out    = (float*)d_out;
    float* pw_out = out + (size_t)nTok * D_;      // prompt_weights output section

    float* wsf = (float*)d_ws;
    float* Kt  = wsf;                              // 3*10*256 floats
    float* Vt  = Kt + 3 * PLEN_ * D_;              // 3*10*256 floats

    const float* wq = ipw;                         // in_proj rows 0..255
    const float* bq = ipb;

    const size_t smem_bytes = (size_t)SMEM_FLOATS * sizeof(float); // ~105.7 KB (<320 KB/WGP)

    k_prep <<<30,   256, 0,          stream>>>(cp, wp, lp, ipw, ipb, Kt, Vt);
    k_fused<<<nBlk, 256, smem_bytes, stream>>>(q, w1, b1, w2, b2, wq, bq, ow, ob,
                                               lng, lnb, Kt, Vt, out, pw_out);
}